// PCACGenerator_77498389889547
// MI455X (gfx1250) — compile-verified
//
#include <hip/hip_runtime.h>

#define BN_EPS 1e-5f

typedef float v2f __attribute__((ext_vector_type(2)));
typedef float v8f __attribute__((ext_vector_type(8)));

// ---------------------------------------------------------------- utilities
__global__ void zero_stats_kernel(float* stats) {
  // 128 threads: stats[0..63]=sum, stats[64..127]=sumsq
  stats[threadIdx.x] = 0.0f;
}

// ---------------------------------------------------------------- conv1 + stats
// h = einsum('knc,kcd->nd', gather(feats, nbr)*mask, W1)
//   = G(N x 81) @ W1cat(81 x 64), K padded 81 -> 84 (21 chunks of 4).
// One wave handles a 16-voxel tile; 4 WMMA column tiles of 16 channels.
// - all 27 neighbor indices preloaded per lane (one batched load clause)
// - K loop fully unrolled: kk/3, kk%3 fold to constants; half-wave (off=0/2)
//   selection is a single cndmask between register-resident idx values.
// - LDS weights fragment-interleaved: w1i[s][col][r] = W1cat[4s+r][col]
//   so one B fragment = one contiguous 8B ds_load_b64.
__global__ __launch_bounds__(256)
void conv1_kernel(const float* __restrict__ feats,
                  const float* __restrict__ W1,   // (27,3,64) = (81,64) flat
                  const int*   __restrict__ nbr,  // (27,N)
                  float*       __restrict__ h,    // (N+1,64)
                  float*       __restrict__ stats,// sum[64], sumsq[64]
                  int N) {
  __shared__ float w1i[21 * 64 * 4];   // [s][col][r], rows 81..83 zero
  __shared__ float stat_s[128];

  const int tid = threadIdx.x;
  for (int i = tid; i < 21 * 256; i += 256) {
    int s = i >> 8, rem = i & 255;
    int c = rem >> 2, r = rem & 3;
    int kk = 4 * s + r;
    w1i[i] = (kk < 81) ? W1[kk * 64 + c] : 0.0f;
  }
  if (tid < 128) stat_s[tid] = 0.0f;
  __syncthreads();

  const int lane   = tid & 31;
  const int wave   = tid >> 5;
  const int tile   = blockIdx.x * 8 + wave;
  const int m0     = tile * 16;
  const int col    = lane & 15;
  const bool hi    = (lane & 16) != 0;  // half-wave: K sub-rows {2,3} vs {0,1}
  const int off    = hi ? 2 : 0;
  const int rowsel = hi ? 8 : 0;

  if (m0 < N) {
    const int   mraw    = m0 + col;
    const float rowmask = (mraw < N) ? 1.0f : 0.0f;
    const int   m       = (mraw < N) ? mraw : (N - 1);

    // preload all 27 neighbor indices (independent -> one load clause)
    int idx[27];
    #pragma unroll
    for (int k = 0; k < 27; ++k) idx[k] = nbr[k * N + m];

    v8f acc0 = {}, acc1 = {}, acc2 = {}, acc3 = {};

    #pragma unroll
    for (int s = 0; s < 21; ++s) {
      float a[2];
      #pragma unroll
      for (int e = 0; e < 2; ++e) {
        const int  kkL = 4 * s + e;      // kk for lanes 0..15
        const int  kkH = 4 * s + 2 + e;  // kk for lanes 16..31
        const bool vL  = (kkL < 81);
        const bool vH  = (kkH < 81);
        float val = 0.0f;
        if (vL || vH) {
          int   iL = vL ? idx[kkL / 3] : 0;
          int   iH = vH ? idx[kkH / 3] : 0;
          int   i  = hi ? iH : iL;
          int   c  = hi ? (kkH % 3) : (kkL % 3);
          float em = hi ? (vH ? 1.0f : 0.0f) : (vL ? 1.0f : 0.0f);
          float mk = (i >= 0) ? (rowmask * em) : 0.0f;
          int   si = (i >= 0) ? i : 0;
          val = mk * feats[3 * si + c];
        }
        a[e] = val;
      }
      v2f A = {a[0], a[1]};

      const float* wb = &w1i[(s * 64 + col) * 4 + off];
      v2f B0 = *(const v2f*)(wb);
      v2f B1 = *(const v2f*)(wb + 16 * 4);
      v2f B2 = *(const v2f*)(wb + 32 * 4);
      v2f B3 = *(const v2f*)(wb + 48 * 4);

      acc0 = __builtin_amdgcn_wmma_f32_16x16x4_f32(false, A, false, B0, (short)0, acc0, false, false);
      acc1 = __builtin_amdgcn_wmma_f32_16x16x4_f32(false, A, false, B1, (short)0, acc1, false, false);
      acc2 = __builtin_amdgcn_wmma_f32_16x16x4_f32(false, A, false, B2, (short)0, acc2, false, false);
      acc3 = __builtin_amdgcn_wmma_f32_16x16x4_f32(false, A, false, B3, (short)0, acc3, false, false);
    }

    // store h tile + accumulate channel stats (padding rows are all-zero)
    v8f accs[4] = {acc0, acc1, acc2, acc3};
    #pragma unroll
    for (int t = 0; t < 4; ++t) {
      float s1 = 0.0f, s2 = 0.0f;
      const int ch = 16 * t + col;
      #pragma unroll
      for (int j = 0; j < 8; ++j) {
        float v = accs[t][j];
        s1 += v;
        s2 += v * v;
        int row = m0 + rowsel + j;
        if (row < N) h[(size_t)row * 64 + ch] = v;
      }
      atomicAdd(&stat_s[ch], s1);
      atomicAdd(&stat_s[64 + ch], s2);
    }
  }
  __syncthreads();
  if (tid < 128) atomicAdd(&stats[tid], stat_s[tid]);
}

// ---------------------------------------------------------------- BN finalize
__global__ void bn_finalize_kernel(const float* __restrict__ stats,
                                   const float* __restrict__ gamma,
                                   const float* __restrict__ beta,
                                   float* __restrict__ scsh, int N) {
  int c = threadIdx.x;
  if (c < 64) {
    float inv_n = 1.0f / (float)N;
    float mean  = stats[c] * inv_n;
    float var   = stats[64 + c] * inv_n - mean * mean;
    float sc    = gamma[c] * rsqrtf(var + BN_EPS);
    scsh[c]      = sc;
    scsh[64 + c] = beta[c] - mean * sc;
  }
}

// ---------------------------------------------------------------- normalize
// h <- relu(h*scale + shift) in place (rows 0..N-1); row N <- 0 (null row for
// invalid neighbors). float4 per thread, fully coalesced.
__global__ __launch_bounds__(256)
void normalize_kernel(float* __restrict__ h, const float* __restrict__ scsh, int N) {
  __shared__ float sc_s[64];
  __shared__ float sh_s[64];
  const int tid = threadIdx.x;
  if (tid < 64) { sc_s[tid] = scsh[tid]; sh_s[tid] = scsh[64 + tid]; }
  __syncthreads();

  long g = (long)blockIdx.x * 256 + tid;        // one float4 group
  long total = (long)(N + 1) * 16;
  if (g >= total) return;
  int row = (int)(g >> 4);
  int c4  = ((int)g & 15) * 4;
  float4* p = (float4*)(h + (size_t)row * 64 + c4);
  if (row < N) {
    float4 v = *p;
    v.x = fmaxf(fmaf(v.x, sc_s[c4 + 0], sh_s[c4 + 0]), 0.0f);
    v.y = fmaxf(fmaf(v.y, sc_s[c4 + 1], sh_s[c4 + 1]), 0.0f);
    v.z = fmaxf(fmaf(v.z, sc_s[c4 + 2], sh_s[c4 + 2]), 0.0f);
    v.w = fmaxf(fmaf(v.w, sc_s[c4 + 3], sh_s[c4 + 3]), 0.0f);
    *p = v;
  } else {
    *p = make_float4(0.0f, 0.0f, 0.0f, 0.0f);
  }
}

// ---------------------------------------------------------------- conv2
// out = sum_k mask_t * h'[idx_t[k]] @ W2[k],  idx_t[k] = nbr[26-k],
// h' pre-normalized, invalid neighbors -> zero row h'[N].
// One wave: TWO 16-voxel row tiles (32 voxels) sharing B fragments.
// Inner loop is pure loads + WMMA: 2x global b64 + 1x ds b64 + 2x wmma.
__global__ __launch_bounds__(256)
void conv2_kernel(const float* __restrict__ hp,    // (N+1,64) normalized
                  const float* __restrict__ W2,    // (27,64,3)
                  const int*   __restrict__ nbr,   // (27,N)
                  float*       __restrict__ out,   // (N,3)
                  int N) {
  __shared__ float w2t[82 * 64];   // [k][d][c] transposed, row 81 = zeros

  const int tid = threadIdx.x;
  for (int i = tid; i < 82 * 64; i += 256) {
    float v = 0.0f;
    if (i < 81 * 64) {
      int kd = i >> 6, c = i & 63;
      int k = kd / 3, d = kd - 3 * k;
      v = W2[(k * 64 + c) * 3 + d];
    }
    w2t[i] = v;
  }
  __syncthreads();

  const int lane   = tid & 31;
  const int wave   = tid >> 5;
  const int tile   = blockIdx.x * 8 + wave;   // 32-voxel tile
  const int m0     = tile * 32;
  const int col    = lane & 15;
  const int off    = (lane & 16) ? 2 : 0;
  const int rowsel = (lane & 16) ? 8 : 0;

  if (m0 < N) {
    const int mA = ((m0 + col)      < N) ? (m0 + col)      : (N - 1);
    const int mB = ((m0 + 16 + col) < N) ? (m0 + 16 + col) : (N - 1);
    const bool bsel = (col < 3);

    v8f accA = {}, accB = {};
    for (int k = 0; k < 27; ++k) {
      const int krev = (26 - k) * N;
      const int iA = nbr[krev + mA];
      const int iB = nbr[krev + mB];
      // invalid neighbor -> zero row N (contribution exactly 0)
      const float* hA = hp + (size_t)((iA >= 0) ? iA : N) * 64 + off;
      const float* hB = hp + (size_t)((iB >= 0) ? iB : N) * 64 + off;
      const float* wk = &w2t[(bsel ? (k * 3 + col) : 81) * 64 + off];

      #pragma unroll
      for (int s = 0; s < 16; ++s) {
        v2f A0 = *(const v2f*)(hA + 4 * s);
        v2f A1 = *(const v2f*)(hB + 4 * s);
        v2f B  = *(const v2f*)(wk + 4 * s);
        accA = __builtin_amdgcn_wmma_f32_16x16x4_f32(false, A0, false, B, (short)0, accA, false, false);
        accB = __builtin_amdgcn_wmma_f32_16x16x4_f32(false, A1, false, B, (short)0, accB, false, false);
      }
    }

    if (bsel) {
      #pragma unroll
      for (int j = 0; j < 8; ++j) {
        int rowA = m0 + rowsel + j;
        int rowB = rowA + 16;
        if (rowA < N) out[(size_t)rowA * 3 + col] = accA[j];
        if (rowB < N) out[(size_t)rowB * 3 + col] = accB[j];
      }
    }
  }
}

// ---------------------------------------------------------------- launch
extern "C" void kernel_launch(void* const* d_in, const int* in_sizes, int n_in,
                              void* d_out, int out_size, void* d_ws, size_t ws_size,
                              hipStream_t stream) {
  const float* feats = (const float*)d_in[0];  // (N,3)
  const float* W1    = (const float*)d_in[1];  // (27,3,64)
  const float* gamma = (const float*)d_in[2];  // (64)
  const float* beta  = (const float*)d_in[3];  // (64)
  const float* W2    = (const float*)d_in[4];  // (27,64,3)
  const int*   nbr   = (const int*)d_in[5];    // (27,N)
  float* out = (float*)d_out;                  // (N,3)

  const int N = in_sizes[0] / 3;

  float* stats = (float*)d_ws;      // 128 floats: sum/sumsq
  float* scsh  = stats + 128;       // 128 floats: scale/shift
  float* h     = stats + 256;       // (N+1)*64 floats

  const int tiles1 = (N + 15) / 16;
  const int wgs1   = (tiles1 + 7) / 8;
  const long grp   = (long)(N + 1) * 16;       // float4 groups
  const int wgsn   = (int)((grp + 255) / 256);
  const int tiles2 = (N + 31) / 32;
  const int wgs2   = (tiles2 + 7) / 8;

  zero_stats_kernel<<<1, 128, 0, stream>>>(stats);
  conv1_kernel<<<wgs1, 256, 0, stream>>>(feats, W1, nbr, h, stats, N);
  bn_finalize_kernel<<<1, 64, 0, stream>>>(stats, gamma, beta, scsh, N);
  normalize_kernel<<<wgsn, 256, 0, stream>>>(h, scsh, N);
  conv2_kernel<<<wgs2, 256, 0, stream>>>(h, W2, nbr, out, N);
}